// XEncoder_309237645936
// MI455X (gfx1250) — compile-verified
//
#include <hip/hip_runtime.h>
#include <math.h>

// ---------------------------------------------------------------------------
// CDNA5 (gfx1250) fp32-WMMA implementation of the XEncoder reference.
// All matrix math goes through V_WMMA_F32_16X16X4_F32 (wave32).
// ---------------------------------------------------------------------------

typedef float v2f __attribute__((ext_vector_type(2)));
typedef float v4f __attribute__((ext_vector_type(4)));
typedef float v8f __attribute__((ext_vector_type(8)));

static __device__ __forceinline__ v8f vzero8() {
  v8f z;
#pragma unroll
  for (int i = 0; i < 8; ++i) z[i] = 0.0f;
  return z;
}
static __device__ __forceinline__ v4f vzero4() {
  v4f z;
#pragma unroll
  for (int i = 0; i < 4; ++i) z[i] = 0.0f;
  return z;
}

// D = A(16x4 f32) * B(4x16 f32) + C(16x16 f32), wave32.
static __device__ __forceinline__ v8f wmma4(v2f a, v2f b, v8f c) {
  return __builtin_amdgcn_wmma_f32_16x16x4_f32(
      /*neg_a=*/false, a, /*neg_b=*/false, b,
      /*c_mod=*/(short)0, c, /*reuse_a=*/false, /*reuse_b=*/false);
}

#define T2   2048   // sequence length
#define DK   128    // per-group head dim
#define GRP  16     // H * B = 8 * 2 flat-view groups
#define Dm   1024   // model dim
#define Mrows 4096  // B * T

static __device__ __forceinline__ float gelu_exact(float v) {
  return 0.5f * v * (1.0f + erff(v * 0.70710678118654752440f));
}

// ---------------------------------------------------------------------------
// Generic fp32 WMMA GEMM:  C[M,N] = act( opA(A)[M,K] @ Bw[K,N] + bias (+resid) )
//   - if mu != nullptr: A element gets LayerNorm applied on load:
//         a = (A[m,k]-mu[m])*rstd[m]*lng[k] + lnb[k]
//   - if resid != nullptr: += resid[m*N+n] before activation
//   - act==1 -> exact GELU
//   - Cout (row-major [M,N]) and Ctr (transposed [b, N, Ttr], m = b*Ttr+t)
//     are each optional.
// Tile: 128x128 per workgroup (8 waves, each 64x32 = 4x2 WMMA tiles), Kblk=32.
// ---------------------------------------------------------------------------
__global__ __launch_bounds__(256) void gemm_kernel(
    const float* __restrict__ A, const float* __restrict__ Bw,
    const float* __restrict__ bias, float* __restrict__ Cout,
    float* __restrict__ Ctr, const float* __restrict__ resid,
    const float* __restrict__ mu, const float* __restrict__ rstd,
    const float* __restrict__ lng, const float* __restrict__ lnb,
    int M, int N, int K, int act, int Ttr) {
  __shared__ float As[128][36];   // +4 pad: conflict-free A-fragment reads
  __shared__ float Bs[32][132];   // +4 pad: conflict-free B-fragment reads

  const int tid  = threadIdx.x;
  const int wave = tid >> 5;
  const int lane = tid & 31;
  const int half = lane >> 4;
  const int ln   = lane & 15;
  const int wm   = wave >> 2;     // 0..1  (rows of 64)
  const int wn   = wave & 3;      // 0..3  (cols of 32)
  const int n0   = blockIdx.x * 128;
  const int m0   = blockIdx.y * 128;

  v8f acc[4][2];
#pragma unroll
  for (int mt = 0; mt < 4; ++mt)
#pragma unroll
    for (int nt = 0; nt < 2; ++nt) acc[mt][nt] = vzero8();

  const bool do_ln = (mu != nullptr);

  for (int kk = 0; kk < K; kk += 32) {
    // --- stage A tile [128 x 32] ---
#pragma unroll
    for (int it = 0; it < 4; ++it) {
      int flat = (tid + it * 256) * 4;
      int row  = flat >> 5;
      int col  = flat & 31;
      v4f v = *(const v4f*)(A + (size_t)(m0 + row) * K + kk + col);
      if (do_ln) {
        float mm = mu[m0 + row];
        float rr = rstd[m0 + row];
        v4f gg = *(const v4f*)(lng + kk + col);
        v4f bb = *(const v4f*)(lnb + kk + col);
        v = (v - mm) * rr * gg + bb;
      }
      *(v4f*)&As[row][col] = v;
    }
    // --- stage B tile [32 x 128] ---
#pragma unroll
    for (int it = 0; it < 4; ++it) {
      int flat = (tid + it * 256) * 4;
      int row  = flat >> 7;
      int col  = flat & 127;
      *(v4f*)&Bs[row][col] = *(const v4f*)(Bw + (size_t)(kk + row) * N + n0 + col);
    }
    __syncthreads();

#pragma unroll
    for (int k4 = 0; k4 < 8; ++k4) {
      v2f af[4];
#pragma unroll
      for (int mt = 0; mt < 4; ++mt)
        af[mt] = *(const v2f*)&As[wm * 64 + mt * 16 + ln][k4 * 4 + half * 2];
      v2f bf[2];
#pragma unroll
      for (int nt = 0; nt < 2; ++nt) {
        bf[nt].x = Bs[k4 * 4 + half * 2 + 0][wn * 32 + nt * 16 + ln];
        bf[nt].y = Bs[k4 * 4 + half * 2 + 1][wn * 32 + nt * 16 + ln];
      }
#pragma unroll
      for (int mt = 0; mt < 4; ++mt)
#pragma unroll
        for (int nt = 0; nt < 2; ++nt)
          acc[mt][nt] = wmma4(af[mt], bf[nt], acc[mt][nt]);
    }
    __syncthreads();
  }

  // --- epilogue ---
#pragma unroll
  for (int mt = 0; mt < 4; ++mt)
#pragma unroll
    for (int nt = 0; nt < 2; ++nt)
#pragma unroll
      for (int r = 0; r < 8; ++r) {
        int m = m0 + wm * 64 + mt * 16 + r + 8 * half;   // C layout: M = reg + 8*half
        int n = n0 + wn * 32 + nt * 16 + ln;             // C layout: N = lane&15
        float v = acc[mt][nt][r] + bias[n];
        if (resid) v += resid[(size_t)m * N + n];
        if (act == 1) v = gelu_exact(v);
        if (Cout) Cout[(size_t)m * N + n] = v;
        if (Ctr) {
          int bb = m / Ttr;
          int tt = m - bb * Ttr;
          Ctr[((size_t)bb * N + n) * Ttr + tt] = v;
        }
      }
}

// ---------------------------------------------------------------------------
// Dense-softmax attention (global branch), flash-style online softmax.
// One workgroup = one group g x 128 query rows. 8 waves, 16 rows per wave.
// S = Q K^T * (1/sqrt(1024)) + adjtab[|i-j|];  O = softmax(S) V.
// Both Q.K^T and P.V use V_WMMA_F32_16X16X4_F32.
// ---------------------------------------------------------------------------
__global__ __launch_bounds__(256) void flash_global_kernel(
    const float* __restrict__ QL, const float* __restrict__ KL,
    const float* __restrict__ VL, float* __restrict__ XO,
    const float* __restrict__ adj_wp, const float* __restrict__ adj_bp) {
  constexpr int BK = 64;
  __shared__ float Ks[BK][DK + 4];     // 33 KB, padded for B-fragment reads
  __shared__ float Vs[BK][DK + 4];     // 33 KB
  __shared__ float Ps[8][16][BK + 2];  // 33 KB, per-wave P transpose staging
  __shared__ float AdjT[T2];           // 8 KB, adj depends only on |i-j|

  const int tid  = threadIdx.x;
  const int wave = tid >> 5;
  const int lane = tid & 31;
  const int half = lane >> 4;
  const int ln   = lane & 15;

  const int g     = blockIdx.y;
  const int qrow0 = blockIdx.x * 128 + wave * 16;

  const float aw    = adj_wp[0];
  const float ab    = adj_bp[0];
  const float scale = 0.03125f;  // 1/sqrt(HID=1024)

  // Precompute adj lookup table.
  for (int d = tid; d < T2; d += 256) {
    float fd = (float)d;
    AdjT[d] = expf(-fabsf(aw * fd * fd - ab));
  }
  __syncthreads();

  const float* Qg = QL + (size_t)g * T2 * DK;
  const float* Kg = KL + (size_t)g * T2 * DK;
  const float* Vg = VL + (size_t)g * T2 * DK;
  float*       Og = XO + (size_t)g * T2 * DK;

  // Q in A-fragment layout: lane holds (m = ln, k = k4*4 + 2*half + {0,1}).
  v2f qf[32];
  {
    const float* qrow = Qg + (size_t)(qrow0 + ln) * DK;
#pragma unroll
    for (int c4 = 0; c4 < 32; ++c4)
      qf[c4] = *(const v2f*)(qrow + c4 * 4 + half * 2);
  }

  v8f acc[8];                           // O accumulator, 16x128 (8 n-tiles)
  float mrow[8], lrow[8];
#pragma unroll
  for (int d8 = 0; d8 < 8; ++d8) acc[d8] = vzero8();
#pragma unroll
  for (int r = 0; r < 8; ++r) { mrow[r] = -3.0e38f; lrow[r] = 0.0f; }

  for (int kb = 0; kb < T2; kb += BK) {
    // ---- cooperative K/V block load (coalesced float4) ----
    {
      const v4f* ksrc = (const v4f*)(Kg + (size_t)kb * DK);
      const v4f* vsrc = (const v4f*)(Vg + (size_t)kb * DK);
#pragma unroll
      for (int it = 0; it < 8; ++it) {
        int idx  = tid + it * 256;  // float4 index, 2048 total
        int flat = idx * 4;
        int row  = flat >> 7;
        int col  = flat & 127;
        *(v4f*)&Ks[row][col] = ksrc[idx];
        *(v4f*)&Vs[row][col] = vsrc[idx];
      }
    }
    __syncthreads();

    // ---- S = Q K^T (4 key tiles x 32 K-steps of WMMA) ----
    v8f Sv[4];
#pragma unroll
    for (int nt = 0; nt < 4; ++nt) {
      v8f s = vzero8();
#pragma unroll
      for (int c4 = 0; c4 < 32; ++c4) {
        v2f b = *(const v2f*)&Ks[nt * 16 + ln][c4 * 4 + half * 2];
        s = wmma4(qf[c4], b, s);
      }
      Sv[nt] = s;
    }

    // ---- scale + adj bias, online softmax update ----
#pragma unroll
    for (int r = 0; r < 8; ++r) {
      const int qi = qrow0 + r + 8 * half;
      float mx = -3.0e38f;
#pragma unroll
      for (int nt = 0; nt < 4; ++nt) {
        int kj = kb + nt * 16 + ln;
        int d  = qi - kj; if (d < 0) d = -d;
        float s = Sv[nt][r] * scale + AdjT[d];
        Sv[nt][r] = s;
        mx = fmaxf(mx, s);
      }
#pragma unroll
      for (int off = 1; off < 16; off <<= 1)   // reduce within 16-lane half
        mx = fmaxf(mx, __shfl_xor(mx, off, 32));

      float mnew = fmaxf(mrow[r], mx);
      float sc   = expf(mrow[r] - mnew);
      mrow[r] = mnew;

      float ps = 0.0f;
#pragma unroll
      for (int nt = 0; nt < 4; ++nt) {
        float p = expf(Sv[nt][r] - mnew);
        Sv[nt][r] = p;
        ps += p;
      }
#pragma unroll
      for (int off = 1; off < 16; off <<= 1)
        ps += __shfl_xor(ps, off, 32);
      lrow[r] = lrow[r] * sc + ps;
#pragma unroll
      for (int d8 = 0; d8 < 8; ++d8) acc[d8][r] *= sc;
    }

    // ---- P through per-wave LDS to reach A-fragment layout ----
#pragma unroll
    for (int nt = 0; nt < 4; ++nt)
#pragma unroll
      for (int r = 0; r < 8; ++r)
        Ps[wave][r + 8 * half][nt * 16 + ln] = Sv[nt][r];
    // (LDS ops from the same wave are in-order; compiler inserts ds waits)

    // ---- O += P V  (16 K-steps x 8 d-tiles of WMMA) ----
#pragma unroll
    for (int k4 = 0; k4 < BK / 4; ++k4) {
      v2f af = *(const v2f*)&Ps[wave][ln][k4 * 4 + half * 2];
#pragma unroll
      for (int d8 = 0; d8 < 8; ++d8) {
        v2f bf;
        bf.x = Vs[k4 * 4 + half * 2 + 0][d8 * 16 + ln];
        bf.y = Vs[k4 * 4 + half * 2 + 1][d8 * 16 + ln];
        acc[d8] = wmma4(af, bf, acc[d8]);
      }
    }
    __syncthreads();
  }

  // ---- normalize and store x_g ----
#pragma unroll
  for (int d8 = 0; d8 < 8; ++d8)
#pragma unroll
    for (int r = 0; r < 8; ++r) {
      int m = qrow0 + r + 8 * half;
      Og[(size_t)m * DK + d8 * 16 + ln] = acc[d8][r] / lrow[r];
    }
}

// ---------------------------------------------------------------------------
// Local (windowed) softmax branch + sigmoid blend.
// Window is <=9 keys/query (~150 MFLOP total) -> VALU, one wave per row.
// XO <- a * x_g + (1-a) * x_l,  a = sigmoid(alpha).
// ---------------------------------------------------------------------------
__global__ __launch_bounds__(256) void local_blend_kernel(
    const float* __restrict__ QL, const float* __restrict__ KL,
    const float* __restrict__ VL, float* __restrict__ XO,
    const float* __restrict__ alphap, const float* __restrict__ adj_wp,
    const float* __restrict__ adj_bp) {
  const int wave = threadIdx.x >> 5;
  const int lane = threadIdx.x & 31;
  const int g = blockIdx.y;
  const int i = blockIdx.x * 8 + wave;

  const float* Qg = QL + (size_t)g * T2 * DK;
  const float* Kg = KL + (size_t)g * T2 * DK;
  const float* Vg = VL + (size_t)g * T2 * DK;

  const float a  = 1.0f / (1.0f + expf(-alphap[0]));
  const float aw = adj_wp[0];
  const float ab = adj_bp[0];

  v4f q = *(const v4f*)(Qg + (size_t)i * DK + lane * 4);

  int jlo = i - 4; if (jlo < 0) jlo = 0;
  int jhi = i + 4; if (jhi > T2 - 1) jhi = T2 - 1;
  const int cnt = jhi - jlo + 1;

  float s[9];
  float mx = -3.0e38f;
  for (int jj = 0; jj < cnt; ++jj) {
    int j = jlo + jj;
    v4f k = *(const v4f*)(Kg + (size_t)j * DK + lane * 4);
    float d = q.x * k.x + q.y * k.y + q.z * k.z + q.w * k.w;
#pragma unroll
    for (int off = 1; off < 32; off <<= 1) d += __shfl_xor(d, off, 32);
    float dist = fabsf((float)(i - j));
    float sj = d * 0.03125f + expf(-fabsf(aw * dist * dist - ab));
    s[jj] = sj;
    mx = fmaxf(mx, sj);
  }
  float den = 0.0f;
  for (int jj = 0; jj < cnt; ++jj) {
    s[jj] = expf(s[jj] - mx);
    den += s[jj];
  }
  v4f o = vzero4();
  for (int jj = 0; jj < cnt; ++jj) {
    v4f v = *(const v4f*)(Vg + (size_t)(jlo + jj) * DK + lane * 4);
    o += v * s[jj];
  }
  float* xo = XO + (size_t)g * T2 * DK + (size_t)i * DK + lane * 4;
  v4f xg = *(v4f*)xo;
  *(v4f*)xo = xg * a + o * ((1.0f - a) / den);
}

// ---------------------------------------------------------------------------
// LayerNorm row statistics (mean, 1/sqrt(var+eps)) over D=1024; 1 wave/row.
// ---------------------------------------------------------------------------
__global__ __launch_bounds__(256) void ln_stats_kernel(
    const float* __restrict__ X2, float* __restrict__ mu,
    float* __restrict__ rstd) {
  const int wave = threadIdx.x >> 5;
  const int lane = threadIdx.x & 31;
  const int row  = blockIdx.x * 8 + wave;
  const float* xr = X2 + (size_t)row * Dm;
  float s = 0.0f, ss = 0.0f;
#pragma unroll
  for (int k = 0; k < 32; ++k) {
    float v = xr[lane + k * 32];
    s += v;
    ss += v * v;
  }
#pragma unroll
  for (int off = 1; off < 32; off <<= 1) {
    s  += __shfl_xor(s, off, 32);
    ss += __shfl_xor(ss, off, 32);
  }
  if (lane == 0) {
    float m   = s * (1.0f / (float)Dm);
    float var = ss * (1.0f / (float)Dm) - m * m;
    mu[row]   = m;
    rstd[row] = rsqrtf(var + 1e-5f);
  }
}

// ---------------------------------------------------------------------------
// Host-side orchestration.
// ---------------------------------------------------------------------------
extern "C" void kernel_launch(void* const* d_in, const int* in_sizes, int n_in,
                              void* d_out, int out_size, void* d_ws,
                              size_t ws_size, hipStream_t stream) {
  (void)in_sizes; (void)n_in; (void)out_size; (void)ws_size;
  const float* x    = (const float*)d_in[0];
  // d_in[1] = seq_len: unused by the reference computation.
  const float* Wq   = (const float*)d_in[2];
  const float* bq   = (const float*)d_in[3];
  const float* Wk   = (const float*)d_in[4];
  const float* bk   = (const float*)d_in[5];
  const float* Wv   = (const float*)d_in[6];
  const float* bv   = (const float*)d_in[7];
  const float* Wo   = (const float*)d_in[8];
  const float* bo   = (const float*)d_in[9];
  const float* alpha= (const float*)d_in[10];
  const float* adjw = (const float*)d_in[11];
  const float* adjb = (const float*)d_in[12];
  const float* lng  = (const float*)d_in[13];
  const float* lnb  = (const float*)d_in[14];
  const float* W1   = (const float*)d_in[15];
  const float* b1   = (const float*)d_in[16];
  const float* W2   = (const float*)d_in[17];
  const float* b2   = (const float*)d_in[18];

  float* out = (float*)d_out;          // [x_s (2*512*2048) | x_e (2*512*2048)]
  float* ws  = (float*)d_ws;

  const size_t SLAB = (size_t)Mrows * Dm;   // 4096*1024
  float* QL   = ws;                         // [4096,1024] == [16][2048][128]
  float* KL   = QL + SLAB;
  float* VL   = KL + SLAB;
  float* XO   = VL + SLAB;                  // attention output (blended)
  float* X2   = XO + SLAB;                  // residual + proj
  float* XE   = X2 + SLAB;                  // [4096,512] gelu(W1) row-major
  float* MU   = XE + (size_t)Mrows * 512;
  float* RSTD = MU + Mrows;

  const dim3 blk(256);

  // 1) QKV projections (fp32 WMMA GEMMs).
  gemm_kernel<<<dim3(8, 32), blk, 0, stream>>>(x, Wq, bq, QL, nullptr, nullptr,
      nullptr, nullptr, nullptr, nullptr, Mrows, Dm, Dm, 0, T2);
  gemm_kernel<<<dim3(8, 32), blk, 0, stream>>>(x, Wk, bk, KL, nullptr, nullptr,
      nullptr, nullptr, nullptr, nullptr, Mrows, Dm, Dm, 0, T2);
  gemm_kernel<<<dim3(8, 32), blk, 0, stream>>>(x, Wv, bv, VL, nullptr, nullptr,
      nullptr, nullptr, nullptr, nullptr, Mrows, Dm, Dm, 0, T2);

  // 2) Dense-softmax attention (global branch) -> XO.
  flash_global_kernel<<<dim3(T2 / 128, GRP), blk, 0, stream>>>(
      QL, KL, VL, XO, adjw, adjb);

  // 3) Windowed-softmax branch + sigmoid blend (in place on XO).
  local_blend_kernel<<<dim3(T2 / 8, GRP), blk, 0, stream>>>(
      QL, KL, VL, XO, alpha, adjw, adjb);

  // 4) X2 = x + XO @ Wo + bo.
  gemm_kernel<<<dim3(8, 32), blk, 0, stream>>>(XO, Wo, bo, X2, nullptr, x,
      nullptr, nullptr, nullptr, nullptr, Mrows, Dm, Dm, 0, T2);

  // 5) LayerNorm statistics.
  ln_stats_kernel<<<dim3(Mrows / 8), blk, 0, stream>>>(X2, MU, RSTD);

  // 6) x_e = gelu(LN(X2) @ W1 + b1): row-major to XE, transposed to out[2nd].
  gemm_kernel<<<dim3(4, 32), blk, 0, stream>>>(X2, W1, b1, XE,
      out + (size_t)2 * 512 * T2, nullptr, MU, RSTD, lng, lnb,
      Mrows, 512, Dm, 1, T2);

  // 7) x_s = gelu(XE @ W2 + b2): transposed straight into out[1st].
  gemm_kernel<<<dim3(4, 32), blk, 0, stream>>>(XE, W2, b2, nullptr, out,
      nullptr, nullptr, nullptr, nullptr, nullptr, Mrows, 512, 512, 1, T2);
}